// RNN_57208964382771
// MI455X (gfx1250) — compile-verified
//
#include <hip/hip_runtime.h>
#include <hip/hip_bf16.h>
#include <math.h>

// Problem constants (from reference): B=64, T=2048, H=8, V=1000, O=1000
#define B_ 64
#define T_ 2048
#define H_ 8
#define V_ 1000
#define O_ 1000
#define M_ (B_ * T_)              // 131072 rows into the output GEMM
#define NFULL_ (O_ / 16)          // 62 full column tiles
#define NTILES_ ((O_ + 15) / 16)  // 63 total column tiles

typedef __attribute__((ext_vector_type(2))) float v2f;
typedef __attribute__((ext_vector_type(8))) float v8f;

// ---------------------------------------------------------------------------
// Kernel A: fold embedding + input projection:
//   emb_proj[v][i] = b_ih[i] + sum_h emb[v][h] * W_ih[i][h]
// Only 8000 outputs; trivial.
// ---------------------------------------------------------------------------
__global__ void emb_proj_kernel(const float* __restrict__ emb,
                                const float* __restrict__ W_ih,
                                const float* __restrict__ b_ih,
                                float* __restrict__ emb_proj) {
    int gtid = blockIdx.x * blockDim.x + threadIdx.x;
    if (gtid >= V_ * H_) return;
    int v = gtid >> 3;
    int i = gtid & 7;
    float acc = b_ih[i];
#pragma unroll
    for (int h = 0; h < H_; ++h)
        acc = fmaf(emb[v * H_ + h], W_ih[i * H_ + h], acc);
    emb_proj[v * H_ + i] = acc;
}

// ---------------------------------------------------------------------------
// Kernel B: sequential RNN scan (latency-bound; 64 independent chains).
// 8 lanes per batch: lane i owns h[i]; cross-lane h[j] via __shfl within the
// 8-lane subgroup of the wave32. Native CDNA5 TANH keeps the step short.
// Writes length-masked hidden states to workspace: hs[b][t][i].
// ---------------------------------------------------------------------------
__global__ void rnn_scan_kernel(const int* __restrict__ X,
                                const int* __restrict__ lengths,
                                const float* __restrict__ emb_proj,
                                const float* __restrict__ W_hh,
                                const float* __restrict__ b_hh,
                                float* __restrict__ hs) {
    int gtid = blockIdx.x * blockDim.x + threadIdx.x;
    int b = gtid >> 3;   // batch
    int i = gtid & 7;    // hidden index owned by this lane
    if (b >= B_) return;

    float w[H_];
#pragma unroll
    for (int j = 0; j < H_; ++j) w[j] = W_hh[i * H_ + j];  // row i of W_hh
    const float bh = b_hh[i];
    const int len = lengths[b];

    const int* Xb = X + b * T_;
    float* hb = hs + (size_t)b * T_ * H_;

    float h = 0.0f;
    for (int t = 0; t < T_; ++t) {
        int tok = Xb[t];
        float acc = emb_proj[tok * H_ + i] + bh;
        // acc += sum_j h[j] * W_hh[i][j]; h[j] lives in lane j of this 8-group
#pragma unroll
        for (int j = 0; j < H_; ++j)
            acc = fmaf(__shfl(h, j, 8), w[j], acc);
        h = tanhf(acc);
        hb[t * H_ + i] = (t < len) ? h : 0.0f;   // pad_packed_sequence mask
    }
}

// ---------------------------------------------------------------------------
// Kernel C: output projection  logits = hs(131072x8) @ W_out^T(8x1000) + b_out
// Store-bandwidth bound (524 MB out). Per wave: one 16-row stripe of hs held
// in registers as two 16x4 f32 A-tiles; loop over column tiles, each tile =
// two chained V_WMMA_F32_16X16X4_F32 (K=0..3 then K=4..7). Non-temporal
// stores for the streaming output. Main loop has no divergence (EXEC all-1s
// for WMMA, no per-iteration clamp); the single partial tile (cols 992..999)
// is peeled into a guarded epilogue.
//
// ISA layouts used (CDNA5 ISA):
//   A 16x4 f32 (2 VGPRs): lane<16 -> row=lane,    K = {0,1} in v0,v1
//                         lane>=16-> row=lane-16, K = {2,3} in v0,v1
//   B  4x16 f32 (2 VGPRs): lane<16 -> col=lane,    K = {0,1}
//                          lane>=16-> col=lane-16, K = {2,3}
//   D 16x16 f32 (8 VGPRs): v[i]: lanes<16 -> (row=i,   col=lane)
//                                lanes>=16-> (row=i+8, col=lane-16)
// ---------------------------------------------------------------------------
__global__ void out_proj_wmma_kernel(const float* __restrict__ hs,
                                     const float* __restrict__ W_out,
                                     const float* __restrict__ b_out,
                                     float* __restrict__ out) {
    const int wave = (blockIdx.x * blockDim.x + threadIdx.x) >> 5;
    const int lane = threadIdx.x & 31;
    const int l16  = lane & 15;
    const int hi   = lane >> 4;          // lane half selects K pair / row half
    const int row_base = wave * 16;

    // A tiles: each lane grabs 2 consecutive K values of its row (float2)
    const float* arow = hs + (size_t)(row_base + l16) * H_ + hi * 2;
    v2f a0 = *(const v2f*)(arow);        // K = 0..3 tile
    v2f a1 = *(const v2f*)(arow + 4);    // K = 4..7 tile

    // Per-lane base pointer for stores: row (row_base + hi*8), col l16
    float* orow = out + (size_t)(row_base + hi * 8) * O_ + l16;

    // ---- 62 full tiles: no clamp, no predication ----
#pragma unroll 1
    for (int nt = 0; nt < NFULL_; ++nt) {
        int n = nt * 16 + l16;
        const float* brow = W_out + (size_t)n * H_ + hi * 2;   // W_out[n][k..]
        v2f b0 = *(const v2f*)(brow);
        v2f b1 = *(const v2f*)(brow + 4);
        float bias = b_out[n];           // hoisted: overlaps the WMMAs

        v8f c = {};
        c = __builtin_amdgcn_wmma_f32_16x16x4_f32(false, a0, false, b0,
                                                  (short)0, c, false, false);
        c = __builtin_amdgcn_wmma_f32_16x16x4_f32(false, a1, false, b1,
                                                  (short)0, c, false, false);

#pragma unroll
        for (int i = 0; i < 8; ++i)
            __builtin_nontemporal_store(c[i] + bias,
                                        &orow[(size_t)i * O_ + nt * 16]);
    }

    // ---- epilogue: partial tile, cols 992..999 valid ----
    {
        int n  = NFULL_ * 16 + l16;
        int nc = (n < O_) ? n : (O_ - 1);            // clamp loads
        const float* brow = W_out + (size_t)nc * H_ + hi * 2;
        v2f b0 = *(const v2f*)(brow);
        v2f b1 = *(const v2f*)(brow + 4);
        float bias = b_out[nc];

        v8f c = {};
        c = __builtin_amdgcn_wmma_f32_16x16x4_f32(false, a0, false, b0,
                                                  (short)0, c, false, false);
        c = __builtin_amdgcn_wmma_f32_16x16x4_f32(false, a1, false, b1,
                                                  (short)0, c, false, false);

        if (n < O_) {                                // predicate stores only
#pragma unroll
            for (int i = 0; i < 8; ++i)
                __builtin_nontemporal_store(c[i] + bias,
                                            &orow[(size_t)i * O_ + NFULL_ * 16]);
        }
    }
}

// ---------------------------------------------------------------------------
// Kernel D: second tuple output = lengths (int32 bits appended after logits)
// ---------------------------------------------------------------------------
__global__ void tail_lengths_kernel(const int* __restrict__ lengths,
                                    int* __restrict__ out_tail, int count) {
    int t = blockIdx.x * blockDim.x + threadIdx.x;
    if (t < count) out_tail[t] = lengths[t];
}

extern "C" void kernel_launch(void* const* d_in, const int* in_sizes, int n_in,
                              void* d_out, int out_size, void* d_ws, size_t ws_size,
                              hipStream_t stream) {
    const int*   X       = (const int*)d_in[0];
    const int*   lengths = (const int*)d_in[1];
    const float* emb     = (const float*)d_in[2];
    const float* W_ih    = (const float*)d_in[3];
    const float* W_hh    = (const float*)d_in[4];
    const float* b_ih    = (const float*)d_in[5];
    const float* b_hh    = (const float*)d_in[6];
    const float* W_out   = (const float*)d_in[7];
    const float* b_out   = (const float*)d_in[8];

    float* emb_proj = (float*)d_ws;          // 1000*8 f32 = 32 KB
    float* hs       = emb_proj + V_ * H_;    // 64*2048*8 f32 = 4.19 MB
    float* logits   = (float*)d_out;

    // A: fold embedding + input projection
    emb_proj_kernel<<<(V_ * H_ + 255) / 256, 256, 0, stream>>>(
        emb, W_ih, b_ih, emb_proj);

    // B: serial scan, 64 batches x 8 lanes = 512 threads
    rnn_scan_kernel<<<2, 256, 0, stream>>>(
        X, lengths, emb_proj, W_hh, b_hh, hs);

    // C: WMMA output projection. 131072/16 = 8192 waves; 8 waves/block.
    out_proj_wmma_kernel<<<(M_ / 16) / 8, 256, 0, stream>>>(
        hs, W_out, b_out, logits);

    // D: append lengths after the logits (tuple output), if space provided
    int tail = out_size - M_ * O_;
    if (tail > 0) {
        int cnt = (tail < B_) ? tail : B_;
        tail_lengths_kernel<<<1, 64, 0, stream>>>(
            lengths, (int*)logits + (size_t)M_ * O_, cnt);
    }
}